// MDRNN_2233382994435
// MI455X (gfx1250) — compile-verified
//
#include <hip/hip_runtime.h>
#include <cmath>

typedef __attribute__((ext_vector_type(2))) float v2f;
typedef __attribute__((ext_vector_type(8))) float v8f;

#define UNITS     2048
#define INPUT_DIM 512
#define T_STEPS   8192

#define NBLK          32      // persistent blocks for the scan
#define ROWS_PER_BLK  64      // 2048 / 32
#define ROWS_PER_WAVE 8       // 64 rows / 8 waves

// ---------------------------------------------------------------------------
// Phase 0: init workspace (state ping-pong buffers + grid-barrier words).
// Must run every call: harness does not re-poison/reset between replays.
// ---------------------------------------------------------------------------
__global__ void scan_init(float* __restrict__ abuf, unsigned* __restrict__ bar) {
    for (int i = threadIdx.x; i < 2 * UNITS; i += blockDim.x) abuf[i] = 0.0f;
    if (threadIdx.x < 2) bar[threadIdx.x] = 0u;
}

// ---------------------------------------------------------------------------
// Phase 1: xin[t,u] = sum_d x[t,d]*wax[u,d] + ba[u], via V_WMMA_F32_16X16X4_F32.
// One wave owns one 16(t) x 16(u) output tile; K loop steps by 4.
//   A (16x4 of x):   lane L: M = L&15, VGPR0/1 = K = k0 + (L>>4)*2 + {0,1}
//   B (4x16 of waxT):lane L: N = L&15, VGPR0/1 = K = k0 + (L>>4)*2 + {0,1}
//   D: VGPR v: row = t0 + v + (L>>4)*8, col = u0 + (L&15)
// ---------------------------------------------------------------------------
__global__ __launch_bounds__(256)
void xin_gemm_wmma(const float* __restrict__ x,    // [T, 512]
                   const float* __restrict__ wax,  // [2048, 512]
                   const float* __restrict__ ba,   // [2048]
                   float* __restrict__ xin)        // [T, 2048]  (== d_out)
{
    const int lane = threadIdx.x & 31;
    const int wid  = blockIdx.x * 8 + (threadIdx.x >> 5);  // 65536 waves total
    const int t0   = (wid >> 7) * 16;                      // 512 t-tiles
    const int u0   = (wid & 127) * 16;                     // 128 u-tiles

    const int m   = lane & 15;
    const int sel = lane >> 4;                             // 0 or 1

    const float* xrow = x   + (size_t)(t0 + m) * INPUT_DIM;
    const float* wrow = wax + (size_t)(u0 + m) * INPUT_DIM;

    v8f acc = {};
    #pragma unroll 4
    for (int k0 = 0; k0 < INPUT_DIM; k0 += 4) {
        const int kb = k0 + sel * 2;                       // 8B-aligned
        v2f a = *(const v2f*)(xrow + kb);
        v2f b = *(const v2f*)(wrow + kb);
        acc = __builtin_amdgcn_wmma_f32_16x16x4_f32(
                  /*neg_a=*/false, a, /*neg_b=*/false, b,
                  /*c_mod=*/(short)0, acc, /*reuse_a=*/false, /*reuse_b=*/false);
    }

    const int   col  = u0 + m;
    const float bias = ba[col];
    #pragma unroll
    for (int v = 0; v < 8; ++v) {
        const int row = t0 + v + sel * 8;
        xin[(size_t)row * UNITS + col] = acc[v] + bias;
    }
}

// ---------------------------------------------------------------------------
// Phase 2: persistent recurrent scan.  a_{t+1} = tanh(waa @ a_t + xin_t).
// 32 blocks x 256 threads; each wave owns 8 rows, lanes split K (coalesced).
// State ping-pongs between abuf[0..2047] and abuf[2048..4095] each step,
// separated by an atomic grid barrier through L2.
// xin is read from y[t*2048+row] and overwritten in place with ys.
// ---------------------------------------------------------------------------
__global__ __launch_bounds__(256)
void rnn_scan(const float* __restrict__ waa,   // [2048, 2048], resident in L2
              float* __restrict__ y,           // [T, 2048]: xin in, ys out
              float* __restrict__ abuf,        // [2, 2048] state ping-pong
              unsigned* __restrict__ bar)      // [0]=arrive, [1]=generation
{
    __shared__ float a_lds[UNITS];

    const int tid  = threadIdx.x;
    const int lane = tid & 31;
    const int wave = tid >> 5;                                  // 0..7
    const int row0 = blockIdx.x * ROWS_PER_BLK + wave * ROWS_PER_WAVE;

    unsigned gen = 0;

    for (int t = 0; t < T_STEPS; ++t) {
        const float* acur = abuf + (size_t)(t & 1) * UNITS;
        float*       anew = abuf + (size_t)((t + 1) & 1) * UNITS;

        // Stage current state into LDS (2 x float4 per thread).
        {
            const float4* src = (const float4*)acur;
            float4*       dst = (float4*)a_lds;
            dst[tid]       = src[tid];
            dst[tid + 256] = src[tid + 256];
        }
        __syncthreads();

        // 8 rows per wave; lane L covers columns k+L — fully coalesced waa reads.
        float s[ROWS_PER_WAVE];
        #pragma unroll
        for (int r = 0; r < ROWS_PER_WAVE; ++r) s[r] = 0.0f;

        const float* wbase = waa + (size_t)row0 * UNITS + lane;
        #pragma unroll 4
        for (int k = 0; k < UNITS; k += 32) {
            const float aval = a_lds[k + lane];
            #pragma unroll
            for (int r = 0; r < ROWS_PER_WAVE; ++r)
                s[r] = fmaf(wbase[(size_t)r * UNITS + k], aval, s[r]);
        }

        // Full wave32 reduction per row; lane 0 applies tanh and writes.
        #pragma unroll
        for (int r = 0; r < ROWS_PER_WAVE; ++r) {
            float v = s[r];
            v += __shfl_xor(v, 16);
            v += __shfl_xor(v, 8);
            v += __shfl_xor(v, 4);
            v += __shfl_xor(v, 2);
            v += __shfl_xor(v, 1);
            if (lane == 0) {
                const size_t idx = (size_t)t * UNITS + (row0 + r);
                const float out = tanhf(v + y[idx]);   // y[idx] holds xin[t,row]
                y[idx]          = out;                 // overwrite with ys[t,row]
                anew[row0 + r]  = out;
            }
        }

        // ---- device-wide barrier (all state writes visible before next step)
        __threadfence();
        __syncthreads();
        if (tid == 0) {
            const unsigned arrived = atomicAdd(&bar[0], 1u);
            if (arrived == NBLK - 1) {
                atomicExch(&bar[0], 0u);
                __threadfence();
                atomicAdd(&bar[1], 1u);
            } else {
                while (__hip_atomic_load(&bar[1], __ATOMIC_RELAXED,
                                         __HIP_MEMORY_SCOPE_AGENT) <= gen) {
                    __builtin_amdgcn_s_sleep(1);
                }
            }
            ++gen;
        }
        __syncthreads();
        __threadfence();   // acquire side: drop stale lines before reading anew
    }
}

// ---------------------------------------------------------------------------
extern "C" void kernel_launch(void* const* d_in, const int* in_sizes, int n_in,
                              void* d_out, int out_size, void* d_ws, size_t ws_size,
                              hipStream_t stream) {
    const float* x   = (const float*)d_in[0];  // (1, 8192, 512)
    const float* waa = (const float*)d_in[1];  // (2048, 2048)
    const float* wax = (const float*)d_in[2];  // (2048, 512)
    const float* ba  = (const float*)d_in[3];  // (2048, 1)
    float* y = (float*)d_out;                  // (1, 8192, 2048)

    float*    abuf = (float*)d_ws;                              // 2*2048 floats
    unsigned* bar  = (unsigned*)((char*)d_ws + 2 * UNITS * sizeof(float));

    scan_init<<<1, 256, 0, stream>>>(abuf, bar);

    // 512 t-tiles * 128 u-tiles = 65536 waves, 8 waves per block.
    xin_gemm_wmma<<<8192, 256, 0, stream>>>(x, wax, ba, y);

    rnn_scan<<<NBLK, 256, 0, stream>>>(waa, y, abuf, bar);
}